// learner_59794534695534
// MI455X (gfx1250) — compile-verified
//
#include <hip/hip_runtime.h>

// ---------------------------------------------------------------------------
// V-trace / PPO learner loss for MI455X (gfx1250, wave32).
//
// Shapes: T=512, B=8192, A=2.  Bandwidth-bound: 36 B/elem * 4M elems = 151 MB
// per full read pass (fits the 192 MB L2, so the 2nd pass and graph replays
// mostly hit L2).  Roofline floor ~6.5us @ 23.3 TB/s for one pass.
//
// Strategy:
//  * V-trace recurrence is affine in the carry: vtrace_t = a_t + b_t*carry.
//    Chunked affine parallel scan over T (32 chunks x 16 steps):
//      pass A composes per-chunk affines, combine chains them per column,
//      pass B re-scans each chunk with its known entering carry.
//    => 4096 waves per pass instead of 256 for the naive scan.
//  * Two batch columns per thread: float4/float2/int2 loads (half the vmem
//    instructions per byte) and two independent recurrence chains for ILP.
//  * Wave reduction via V_WMMA_F32_16X16X4_F32 (A = partials in K-slot 0,
//    B = all-ones => D[m,n] = partial(m)+partial(m+16)); fixed-order block +
//    final sums, no atomics => bit-deterministic across graph replays.
// ---------------------------------------------------------------------------

#define TT       512
#define BB       8192
#define HB       (BB / 2)          // 4096 column pairs
#define NC       32                // chunks along T
#define CL       (TT / NC)         // 16 timesteps per chunk
#define NTHREADS (NC * HB)         // 131072 threads per pass
#define BLK      256
#define NBLKP    (NTHREADS / BLK)  // 512 blocks in pass A / pass B

#define GAMMA_F  0.99f
#define CLIP_LO  0.8f
#define CLIP_HI  1.2f

typedef __attribute__((ext_vector_type(2))) float v2f;
typedef __attribute__((ext_vector_type(8))) float v8f;

// ratio = exp(logp - logp_act) == p_a*sum(ap) / (sum(p)*ap_a)
__device__ __forceinline__ float importance_ratio(float px, float py,
                                                  float qx, float qy, int a) {
    const float pa = a ? py : px;
    const float qa = a ? qy : qx;
    return (pa * (qx + qy)) / ((px + py) * qa);
}

// Per-element affine coefficients: vtrace_t = at + bt*carry
__device__ __forceinline__ void step_coeffs(float vv, float nvv, float rr,
                                            int a, int d, float px, float py,
                                            float qx, float qy,
                                            float& ratio, float& rho,
                                            float& ndf, float& bt, float& at) {
    ratio = importance_ratio(px, py, qx, qy, a);
    rho   = fminf(ratio, 1.0f);            // rho == c (both thresholds 1.0)
    ndf   = 1.0f - (float)d;
    const float gnd   = GAMMA_F * ndf;
    bt    = rho * gnd;                     // gamma*c*(1-done)
    const float delta = rho * fmaf(gnd, nvv, rr - vv);
    at    = vv + delta - bt * nvv;
}

// ---------------------------------------------------------------------------
// Pass A: per (chunk, column-pair) compose the backward affine of the chunk.
// ---------------------------------------------------------------------------
__global__ __launch_bounds__(BLK) void vtrace_passA(
    const float4* __restrict__ prob4, const float4* __restrict__ aprob4,
    const float2* __restrict__ v2,   const float2* __restrict__ nv2,
    const float2* __restrict__ rw2,  const int2* __restrict__ act2,
    const int2* __restrict__ dn2,
    float2* __restrict__ chA2, float2* __restrict__ chB2)
{
    const int tid = blockIdx.x * BLK + threadIdx.x;
    const int bp  = tid & (HB - 1);        // column pair: columns 2bp, 2bp+1
    const int c   = tid >> 12;             // HB == 2^12
    const int t0  = c * CL;

    float A0 = 0.0f, B0 = 1.0f, A1 = 0.0f, B1 = 1.0f;
    #pragma unroll 4
    for (int t = t0 + CL - 1; t >= t0; --t) {
        const int h = t * HB + bp;         // float2/int2/float4 index, coalesced
        const float2 vv  = v2[h];
        const float2 nvv = nv2[h];
        const float2 rr  = rw2[h];
        const int2   aa  = act2[h];
        const int2   dd  = dn2[h];
        const float4 p   = prob4[h];
        const float4 q   = aprob4[h];

        float ratio, rho, ndf, bt, at;
        step_coeffs(vv.x, nvv.x, rr.x, aa.x, dd.x, p.x, p.y, q.x, q.y,
                    ratio, rho, ndf, bt, at);
        A0 = fmaf(bt, A0, at);  B0 = bt * B0;

        step_coeffs(vv.y, nvv.y, rr.y, aa.y, dd.y, p.z, p.w, q.z, q.w,
                    ratio, rho, ndf, bt, at);
        A1 = fmaf(bt, A1, at);  B1 = bt * B1;
    }
    chA2[tid] = make_float2(A0, A1);       // == chA[c*BB + 2bp .. +1]
    chB2[tid] = make_float2(B0, B1);
}

// ---------------------------------------------------------------------------
// Combine: per column, chain NC chunk affines from t=T-1 downward to get the
// carry entering the top of every chunk.
// ---------------------------------------------------------------------------
__global__ __launch_bounds__(BLK) void vtrace_combine(
    const float* __restrict__ nv, const float* __restrict__ chA,
    const float* __restrict__ chB, float* __restrict__ yin)
{
    const int b = blockIdx.x * BLK + threadIdx.x;         // 0..BB-1
    float y = nv[(TT - 1) * BB + b];                      // next_v[-1]
    #pragma unroll
    for (int c = NC - 1; c >= 0; --c) {
        const int i = c * BB + b;
        yin[i] = y;                                       // carry entering chunk c
        y = fmaf(chB[i], y, chA[i]);                      // carry leaving chunk
    }
}

// ---------------------------------------------------------------------------
// Pass B: re-scan each chunk with its known entering carry, accumulate
// critic = sum (v - vtrace)^2 and actor = sum min(surr1, surr2).
// ---------------------------------------------------------------------------
__global__ __launch_bounds__(BLK) void vtrace_passB(
    const float4* __restrict__ prob4, const float4* __restrict__ aprob4,
    const float2* __restrict__ v2,   const float2* __restrict__ nv2,
    const float2* __restrict__ rw2,  const int2* __restrict__ act2,
    const int2* __restrict__ dn2,    const float2* __restrict__ yin2,
    float* __restrict__ blkC, float* __restrict__ blkA)
{
    __shared__ float red[(BLK / 32) * 4];

    const int tid = blockIdx.x * BLK + threadIdx.x;
    const int bp  = tid & (HB - 1);
    const int c   = tid >> 12;
    const int t0  = c * CL;

    const float2 y0 = yin2[tid];
    float carry0 = y0.x, carry1 = y0.y;
    float critic = 0.0f, actor = 0.0f;

    #pragma unroll 4
    for (int t = t0 + CL - 1; t >= t0; --t) {
        const int h = t * HB + bp;
        const float2 vv  = v2[h];
        const float2 nvv = nv2[h];
        const float2 rr  = rw2[h];
        const int2   aa  = act2[h];
        const int2   dd  = dn2[h];
        const float4 p   = prob4[h];
        const float4 q   = aprob4[h];

        {   // column 2bp
            float ratio, rho, ndf, bt, at;
            step_coeffs(vv.x, nvv.x, rr.x, aa.x, dd.x, p.x, p.y, q.x, q.y,
                        ratio, rho, ndf, bt, at);
            const float vt   = fmaf(bt, carry0, at);
            const float adv  = rho * fmaf(GAMMA_F * carry0, ndf, rr.x - vv.x);
            const float diff = vv.x - vt;
            critic = fmaf(diff, diff, critic);
            const float rcl = fminf(fmaxf(ratio, CLIP_LO), CLIP_HI);
            actor += fminf(ratio * adv, rcl * adv);
            carry0 = vt;
        }
        {   // column 2bp+1
            float ratio, rho, ndf, bt, at;
            step_coeffs(vv.y, nvv.y, rr.y, aa.y, dd.y, p.z, p.w, q.z, q.w,
                        ratio, rho, ndf, bt, at);
            const float vt   = fmaf(bt, carry1, at);
            const float adv  = rho * fmaf(GAMMA_F * carry1, ndf, rr.y - vv.y);
            const float diff = vv.y - vt;
            critic = fmaf(diff, diff, critic);
            const float rcl = fminf(fmaxf(ratio, CLIP_LO), CLIP_HI);
            actor += fminf(ratio * adv, rcl * adv);
            carry1 = vt;
        }
    }

    const int lane = threadIdx.x & 31;
    const int w    = threadIdx.x >> 5;
    if (threadIdx.x < (BLK / 32) * 4) red[threadIdx.x] = 0.0f;
    __syncthreads();

#if __has_builtin(__builtin_amdgcn_wmma_f32_16x16x4_f32)
    // EXEC all-ones here (uniform control flow) as WMMA requires.
    v2f av; av.x = critic; av.y = 0.0f;
    v2f bv; bv.x = 1.0f;   bv.y = 1.0f;   // all-ones B (layout-agnostic)
    v8f cz = {0.f, 0.f, 0.f, 0.f, 0.f, 0.f, 0.f, 0.f};
    v8f d1 = __builtin_amdgcn_wmma_f32_16x16x4_f32(false, av, false, bv,
                                                   (short)0, cz, false, false);
    av.x = actor;
    v8f d2 = __builtin_amdgcn_wmma_f32_16x16x4_f32(false, av, false, bv,
                                                   (short)0, cz, false, false);
    // lane 0 holds rows 0-7 of column 0; lane 16 holds rows 8-15.
    float s1 = d1[0]+d1[1]+d1[2]+d1[3]+d1[4]+d1[5]+d1[6]+d1[7];
    float s2 = d2[0]+d2[1]+d2[2]+d2[3]+d2[4]+d2[5]+d2[6]+d2[7];
    if (lane == 0)       { red[w*4 + 0] = s1; red[w*4 + 2] = s2; }
    else if (lane == 16) { red[w*4 + 1] = s1; red[w*4 + 3] = s2; }
#else
    float s1 = critic, s2 = actor;
    for (int o = 16; o > 0; o >>= 1) {
        s1 += __shfl_down(s1, o);
        s2 += __shfl_down(s2, o);
    }
    if (lane == 0) { red[w*4 + 0] = s1; red[w*4 + 2] = s2; }
#endif
    __syncthreads();

    if (threadIdx.x == 0) {
        float cs = 0.0f, as = 0.0f;
        #pragma unroll
        for (int i = 0; i < BLK / 32; ++i) {
            cs += red[4*i + 0] + red[4*i + 1];
            as += red[4*i + 2] + red[4*i + 3];
        }
        blkC[blockIdx.x] = cs;
        blkA[blockIdx.x] = as;
    }
}

// ---------------------------------------------------------------------------
// Finalize: deterministic fixed-order sum of block partials -> scalar loss.
// ---------------------------------------------------------------------------
__global__ void vtrace_final(const float* __restrict__ blkC,
                             const float* __restrict__ blkA,
                             float* __restrict__ out)
{
    if (threadIdx.x == 0 && blockIdx.x == 0) {
        float cs = 0.0f, as = 0.0f;
        for (int i = 0; i < NBLKP; ++i) { cs += blkC[i]; as += blkA[i]; }
        const float invN = 1.0f / (float)(TT * BB);
        // total = actor_loss + critic_loss = -as/N + 0.5*cs/N
        out[0] = (0.5f * cs - as) * invN;
    }
}

// ---------------------------------------------------------------------------
extern "C" void kernel_launch(void* const* d_in, const int* in_sizes, int n_in,
                              void* d_out, int out_size, void* d_ws, size_t ws_size,
                              hipStream_t stream) {
    const float* prob  = (const float*)d_in[0];   // [T,B,2]
    const float* aprob = (const float*)d_in[1];   // [T,B,2]
    const float* v     = (const float*)d_in[2];   // [T,B]
    const float* nv    = (const float*)d_in[3];   // [T,B]
    const float* rw    = (const float*)d_in[4];   // [T,B]
    const int*   act   = (const int*)d_in[5];     // [T,B]
    const int*   dn    = (const int*)d_in[6];     // [T,B]
    float* out = (float*)d_out;

    float* wsf  = (float*)d_ws;                   // ~3.1 MB used
    float* chA  = wsf;                            // NC*BB floats
    float* chB  = chA + NC * BB;                  // NC*BB floats
    float* yin  = chB + NC * BB;                  // NC*BB floats
    float* blkC = yin + NC * BB;                  // NBLKP floats
    float* blkA = blkC + NBLKP;                   // NBLKP floats

    vtrace_passA<<<NBLKP, BLK, 0, stream>>>(
        (const float4*)prob, (const float4*)aprob,
        (const float2*)v, (const float2*)nv, (const float2*)rw,
        (const int2*)act, (const int2*)dn,
        (float2*)chA, (float2*)chB);

    vtrace_combine<<<BB / BLK, BLK, 0, stream>>>(nv, chA, chB, yin);

    vtrace_passB<<<NBLKP, BLK, 0, stream>>>(
        (const float4*)prob, (const float4*)aprob,
        (const float2*)v, (const float2*)nv, (const float2*)rw,
        (const int2*)act, (const int2*)dn, (const float2*)yin,
        blkC, blkA);

    vtrace_final<<<1, 32, 0, stream>>>(blkC, blkA, out);
}